// GPCALayer_88476326298061
// MI455X (gfx1250) — compile-verified
//
#include <hip/hip_runtime.h>

// ---------------------------------------------------------------------------
// GPCA layer for MI455X (gfx1250, wave32).
// Strategy: build CSR once (atomics only in setup); 50 power iterations are
// pure L2-resident gathers (whole working set << 192MB L2); label-prop term
// accumulated in LDS per block; final fp32 GEMM uses V_WMMA_F32_16X16X4_F32.
// ---------------------------------------------------------------------------

#define DD 64          // feature dim (in == out)
#define CC 47          // num classes
#define NPOW 50

typedef __attribute__((ext_vector_type(2))) float v2f;
typedef __attribute__((ext_vector_type(8))) float v8f;

// ---------------- setup kernels ----------------

__global__ __launch_bounds__(64) void colsum_kernel(const float* __restrict__ x,
                                                    float* __restrict__ colsum, int n) {
    int d = threadIdx.x;             // 0..63, coalesced across a row
    float local = 0.f;
    for (int i = blockIdx.x; i < n; i += gridDim.x)
        local += x[(size_t)i * DD + d];
    atomicAdd(&colsum[d], local);
}

__global__ __launch_bounds__(256) void xc_kernel(const float* __restrict__ x,
                                                 const float* __restrict__ colsum,
                                                 float* __restrict__ xc, int n) {
    float inv_n = 1.0f / (float)n;
    long total = (long)n * DD;
    for (long idx = blockIdx.x * 256L + threadIdx.x; idx < total; idx += gridDim.x * 256L)
        xc[idx] = x[idx] - colsum[idx & (DD - 1)] * inv_n;
}

__global__ __launch_bounds__(256) void count_rows(const int* __restrict__ ei, int e,
                                                  int* __restrict__ counts) {
    for (int i = blockIdx.x * 256 + threadIdx.x; i < e; i += gridDim.x * 256)
        atomicAdd(&counts[ei[i]], 1);   // ei[0..E) = row indices
}

__global__ __launch_bounds__(256) void classcnt_kernel(const int* __restrict__ y,
                                                       const int* __restrict__ tm,
                                                       float* __restrict__ classcnt, int n) {
    for (int i = blockIdx.x * 256 + threadIdx.x; i < n; i += gridDim.x * 256)
        if (tm[i]) atomicAdd(&classcnt[y[i]], 1.0f);
}

__global__ __launch_bounds__(256) void deg_s_kernel(const int* __restrict__ counts,
                                                    const int* __restrict__ tm,
                                                    const int* __restrict__ y,
                                                    const float* __restrict__ classcnt,
                                                    float* __restrict__ deg_inv,
                                                    float* __restrict__ s, int n) {
    for (int i = blockIdx.x * 256 + threadIdx.x; i < n; i += gridDim.x * 256) {
        deg_inv[i] = 1.0f / (float)(counts[i] + 1);          // +1 self loop
        s[i] = tm[i] ? (1.0f / (classcnt[y[i]] + 1e-8f)) : 0.0f;
    }
}

// 3-phase exclusive scan of counts -> rowptr
__global__ __launch_bounds__(1024) void scan1(const int* __restrict__ counts,
                                              int* __restrict__ incl,
                                              int* __restrict__ bsum, int n) {
    __shared__ int sh[1024];
    int t = threadIdx.x;
    int gi = blockIdx.x * 1024 + t;
    sh[t] = (gi < n) ? counts[gi] : 0;
    __syncthreads();
    for (int off = 1; off < 1024; off <<= 1) {
        int v = (t >= off) ? sh[t - off] : 0;
        __syncthreads();
        sh[t] += v;
        __syncthreads();
    }
    if (gi < n) incl[gi] = sh[t];
    if (t == 1023) bsum[blockIdx.x] = sh[1023];
}

__global__ void scan2(const int* __restrict__ bsum, int* __restrict__ boff, int nb) {
    if (threadIdx.x == 0 && blockIdx.x == 0) {
        int run = 0;
        for (int i = 0; i < nb; ++i) { boff[i] = run; run += bsum[i]; }
    }
}

__global__ __launch_bounds__(256) void scan3(const int* __restrict__ incl,
                                             const int* __restrict__ counts,
                                             const int* __restrict__ boff,
                                             int* __restrict__ rowptr,
                                             int* __restrict__ cursor, int n) {
    for (int i = blockIdx.x * 256 + threadIdx.x; i < n; i += gridDim.x * 256) {
        int v = incl[i] + boff[i >> 10];
        rowptr[i] = v - counts[i];
        cursor[i] = v - counts[i];
        if (i == n - 1) rowptr[n] = v;
    }
}

__global__ __launch_bounds__(256) void fill_csr(const int* __restrict__ ei, int e,
                                                int* __restrict__ cursor,
                                                int* __restrict__ csr_col) {
    for (int i = blockIdx.x * 256 + threadIdx.x; i < e; i += gridDim.x * 256) {
        int r = ei[i];
        int c = ei[e + i];             // second row of edge_index
        int pos = atomicAdd(&cursor[r], 1);
        csr_col[pos] = c;
    }
}

// ---------------- hot loop kernels ----------------

// cs[c][d] = sum over train nodes of class c of v[i][d]; LDS-staged (12KB << 320KB/WGP)
__global__ __launch_bounds__(256) void cs_kernel(const float* __restrict__ vin,
                                                 const int* __restrict__ y,
                                                 const int* __restrict__ tm,
                                                 float* __restrict__ cs, int n) {
    __shared__ float sh[CC * DD];
    for (int t = threadIdx.x; t < CC * DD; t += 256) sh[t] = 0.f;
    __syncthreads();
    long total = (long)n * DD;
    for (long idx = blockIdx.x * 256L + threadIdx.x; idx < total; idx += gridDim.x * 256L) {
        int i = (int)(idx >> 6);
        int d = (int)(idx & (DD - 1));
        if (tm[i]) atomicAdd(&sh[y[i] * DD + d], vin[idx]);   // ds_add_f32
    }
    __syncthreads();
    for (int t = threadIdx.x; t < CC * DD; t += 256) {
        float v = sh[t];
        if (v != 0.f) atomicAdd(&cs[t], v);
    }
}

// v_out[i] = 0.45*deg_inv[i]*(v[i] + sum_{j in adj(i)} v[j]) + 0.05*s[i]*cs[y[i]] + 0.5*xc[i]
// One wave per row; lane owns float2 of the 64-wide row => 256B coalesced gathers.
__global__ __launch_bounds__(256) void spmm_combine(const float* __restrict__ vin,
                                                    const float* __restrict__ xc,
                                                    const float* __restrict__ cs,
                                                    const float* __restrict__ deg_inv,
                                                    const float* __restrict__ s,
                                                    const int* __restrict__ y,
                                                    const int* __restrict__ rowptr,
                                                    const int* __restrict__ csr_col,
                                                    float* __restrict__ vout, int n) {
    int wave = (int)((blockIdx.x * 256 + threadIdx.x) >> 5);
    int lane = threadIdx.x & 31;
    if (wave >= n) return;
    int i = wave;
    const float2* vin2 = (const float2*)vin;
    float2 acc = vin2[(size_t)i * 32 + lane];          // self loop
    int beg = rowptr[i], end = rowptr[i + 1];
    int e = beg;
    // 2-deep manual pipeline for L2 latency hiding
    for (; e + 1 < end; e += 2) {
        int j0 = csr_col[e];
        int j1 = csr_col[e + 1];
        float2 t0 = vin2[(size_t)j0 * 32 + lane];
        float2 t1 = vin2[(size_t)j1 * 32 + lane];
        acc.x += t0.x + t1.x;
        acc.y += t0.y + t1.y;
    }
    if (e < end) {
        float2 t = vin2[(size_t)csr_col[e] * 32 + lane];
        acc.x += t.x; acc.y += t.y;
    }
    float di = deg_inv[i];
    float si = s[i];
    float2 c2 = ((const float2*)cs)[(size_t)y[i] * 32 + lane];
    float2 x2 = ((const float2*)xc)[(size_t)i * 32 + lane];
    float2 o;
    o.x = 0.45f * di * acc.x + 0.05f * si * c2.x + 0.5f * x2.x;
    o.y = 0.45f * di * acc.y + 0.05f * si * c2.y + 0.5f * x2.y;
    ((float2*)vout)[(size_t)i * 32 + lane] = o;
}

// ---------------- final GEMM: out = A[N x 64] @ W[64 x 64] + bias, fp32 WMMA ----------------
// One wave per 16-row tile; 16 K-steps x 4 N-tiles of V_WMMA_F32_16X16X4_F32.
__global__ __launch_bounds__(256) void gemm_wmma(const float* __restrict__ A,
                                                 const float* __restrict__ W,
                                                 const float* __restrict__ bias,
                                                 float* __restrict__ out, int n) {
    int wave = (int)((blockIdx.x * 256 + threadIdx.x) >> 5);
    int lane = threadIdx.x & 31;
    int ntiles = n >> 4;
    if (wave >= ntiles) return;
    int row0 = wave << 4;
    int m  = lane & 15;                  // M (for A) / N (for B) within tile
    int hi = lane >> 4;                  // 0: K pair {0,1}; 1: K pair {2,3}
    int kh = hi << 1;

    v8f acc0 = {}, acc1 = {}, acc2 = {}, acc3 = {};
    for (int kk = 0; kk < 16; ++kk) {
        int k0 = kk * 4 + kh;
        // A fragment: 16x4 fp32, lanes 0-15: M=m,K={0,1}; lanes 16-31: M=m,K={2,3}
        v2f a;
        a.x = A[(size_t)(row0 + m) * DD + k0];
        a.y = A[(size_t)(row0 + m) * DD + k0 + 1];
        // B fragments: 4x16 fp32, lanes 0-15: N=m,K={0,1}; lanes 16-31: N=m,K={2,3}
        v2f b0, b1, b2, b3;
        b0.x = W[(size_t)k0 * DD + 0  + m]; b0.y = W[(size_t)(k0 + 1) * DD + 0  + m];
        b1.x = W[(size_t)k0 * DD + 16 + m]; b1.y = W[(size_t)(k0 + 1) * DD + 16 + m];
        b2.x = W[(size_t)k0 * DD + 32 + m]; b2.y = W[(size_t)(k0 + 1) * DD + 32 + m];
        b3.x = W[(size_t)k0 * DD + 48 + m]; b3.y = W[(size_t)(k0 + 1) * DD + 48 + m];
        acc0 = __builtin_amdgcn_wmma_f32_16x16x4_f32(false, a, false, b0, (short)0, acc0, false, false);
        acc1 = __builtin_amdgcn_wmma_f32_16x16x4_f32(false, a, false, b1, (short)0, acc1, false, false);
        acc2 = __builtin_amdgcn_wmma_f32_16x16x4_f32(false, a, false, b2, (short)0, acc2, false, false);
        acc3 = __builtin_amdgcn_wmma_f32_16x16x4_f32(false, a, false, b3, (short)0, acc3, false, false);
    }
    // C/D layout: VGPR r -> lanes 0-15: M=r, N=lane; lanes 16-31: M=8+r, N=lane-16
    int rbase = row0 + (hi ? 8 : 0);
    for (int r = 0; r < 8; ++r) {
        size_t o = (size_t)(rbase + r) * DD;
        out[o + 0  + m] = acc0[r] + bias[0  + m];
        out[o + 16 + m] = acc1[r] + bias[16 + m];
        out[o + 32 + m] = acc2[r] + bias[32 + m];
        out[o + 48 + m] = acc3[r] + bias[48 + m];
    }
}

// scalar tail for N not a multiple of 16 (N=100000 divides exactly; kept for safety)
__global__ __launch_bounds__(64) void gemm_tail(const float* __restrict__ A,
                                                const float* __restrict__ W,
                                                const float* __restrict__ bias,
                                                float* __restrict__ out,
                                                int start, int n) {
    int i = start + blockIdx.x;
    if (i >= n) return;
    int d = threadIdx.x;
    float acc = bias[d];
    for (int k = 0; k < DD; ++k) acc += A[(size_t)i * DD + k] * W[(size_t)k * DD + d];
    out[(size_t)i * DD + d] = acc;
}

// ---------------------------------------------------------------------------

static inline size_t align256(size_t x) { return (x + 255) & ~(size_t)255; }

extern "C" void kernel_launch(void* const* d_in, const int* in_sizes, int n_in,
                              void* d_out, int out_size, void* d_ws, size_t ws_size,
                              hipStream_t stream) {
    const float* x    = (const float*)d_in[0];
    const float* W    = (const float*)d_in[1];
    const float* bias = (const float*)d_in[2];
    const int*   ei   = (const int*)d_in[3];   // [2, E]
    const int*   y    = (const int*)d_in[4];
    const int*   tm   = (const int*)d_in[5];
    const int N = in_sizes[4];
    const int E = in_sizes[3] / 2;

    // workspace carving (~86 MB total)
    char* p = (char*)d_ws;
    auto carve = [&](size_t bytes) -> void* { void* r = (void*)p; p += align256(bytes); return r; };
    float* deg_inv  = (float*)carve((size_t)N * 4);
    float* s        = (float*)carve((size_t)N * 4);
    float* colsum   = (float*)carve(DD * 4);
    float* classcnt = (float*)carve(64 * 4);
    float* cs       = (float*)carve((size_t)CC * DD * 4);
    float* xc       = (float*)carve((size_t)N * DD * 4);
    float* va       = (float*)carve((size_t)N * DD * 4);
    float* vb       = (float*)carve((size_t)N * DD * 4);
    int* counts  = (int*)carve((size_t)N * 4);
    int* incl    = (int*)carve((size_t)N * 4);
    int* bsum    = (int*)carve(1024 * 4);
    int* boff    = (int*)carve(1024 * 4);
    int* rowptr  = (int*)carve(((size_t)N + 1) * 4);
    int* cursor  = (int*)carve((size_t)N * 4);
    int* csr_col = (int*)carve((size_t)E * 4);

    const int GB = 512;  // generic grid

    hipMemsetAsync(colsum, 0, DD * 4, stream);
    hipMemsetAsync(classcnt, 0, 64 * 4, stream);
    hipMemsetAsync(counts, 0, (size_t)N * 4, stream);

    colsum_kernel<<<256, 64, 0, stream>>>(x, colsum, N);
    count_rows<<<GB, 256, 0, stream>>>(ei, E, counts);
    classcnt_kernel<<<GB, 256, 0, stream>>>(y, tm, classcnt, N);

    int nb = (N + 1023) / 1024;
    scan1<<<nb, 1024, 0, stream>>>(counts, incl, bsum, N);
    scan2<<<1, 1, 0, stream>>>(bsum, boff, nb);
    scan3<<<GB, 256, 0, stream>>>(incl, counts, boff, rowptr, cursor, N);
    fill_csr<<<GB, 256, 0, stream>>>(ei, E, cursor, csr_col);

    deg_s_kernel<<<GB, 256, 0, stream>>>(counts, tm, y, classcnt, deg_inv, s, N);
    xc_kernel<<<GB, 256, 0, stream>>>(x, colsum, xc, N);

    // power iteration: vin starts at xc; outputs alternate va/vb
    const float* vin = xc;
    float* vout = va;
    int spmm_blocks = (N + 7) / 8;   // 8 waves per 256-thread block, 1 wave per row
    for (int it = 0; it < NPOW; ++it) {
        hipMemsetAsync(cs, 0, (size_t)CC * DD * 4, stream);
        cs_kernel<<<128, 256, 0, stream>>>(vin, y, tm, cs, N);
        spmm_combine<<<spmm_blocks, 256, 0, stream>>>(vin, xc, cs, deg_inv, s, y,
                                                      rowptr, csr_col, vout, N);
        vin = vout;
        vout = (vout == va) ? vb : va;
    }

    // final GEMM with fp32 WMMA
    int ntiles = N >> 4;
    int gblocks = (ntiles + 7) / 8;
    gemm_wmma<<<gblocks, 256, 0, stream>>>(vin, W, bias, (float*)d_out, N);
    if (N & 15)
        gemm_tail<<<(N & 15), 64, 0, stream>>>(vin, W, bias, (float*)d_out, ntiles << 4, N);
}